// kNN_VC_1571958030557
// MI455X (gfx1250) — compile-verified
//
#include <hip/hip_runtime.h>
#include <hip/hip_bf16.h>
#include <math.h>

#define SRC   2048
#define TGT   50000
#define DIMS  1024
#define SLICES 8
#define SLICE_LEN (TGT / SLICES)   // 6250
#define NEG_BIG (-1.0e30f)
#define MT    32                   // source rows per block/wave

typedef float v2f __attribute__((ext_vector_type(2)));
typedef float v8f __attribute__((ext_vector_type(8)));

// ---------- sorted top-4 insert (descending) ----------
__device__ __forceinline__ void insert4(float v, int id, float tv[4], int ti[4]) {
    bool b0 = v > tv[0], b1 = v > tv[1], b2 = v > tv[2], b3 = v > tv[3];
    if (b3) {
        tv[3] = b2 ? tv[2] : v;                ti[3] = b2 ? ti[2] : id;
        tv[2] = b2 ? (b1 ? tv[1] : v) : tv[2]; ti[2] = b2 ? (b1 ? ti[1] : id) : ti[2];
        tv[1] = b1 ? (b0 ? tv[0] : v) : tv[1]; ti[1] = b1 ? (b0 ? ti[0] : id) : ti[1];
        tv[0] = b0 ? v : tv[0];                ti[0] = b0 ? id : ti[0];
    }
}

// ---------- kernel 1: normalize source rows ----------
__global__ void __launch_bounds__(256) norm_src_kernel(const float* __restrict__ src,
                                                       float* __restrict__ srcn) {
    __shared__ float red[8];
    const int row = blockIdx.x;
    const float4* r4 = (const float4*)(src + (size_t)row * DIMS);
    const float4 v = r4[threadIdx.x];                 // 256 threads x float4 = 1024
    float ss = v.x * v.x + v.y * v.y + v.z * v.z + v.w * v.w;
    for (int off = 16; off > 0; off >>= 1) ss += __shfl_xor(ss, off, 32);
    const int lid = threadIdx.x & 31, wid = threadIdx.x >> 5;
    if (lid == 0) red[wid] = ss;
    __syncthreads();
    float tot = 0.0f;
    #pragma unroll
    for (int i = 0; i < 8; i++) tot += red[i];
    const float sc = 1.0f / fmaxf(sqrtf(tot), 1e-8f);
    float4* o4 = (float4*)(srcn + (size_t)row * DIMS);
    o4[threadIdx.x] = make_float4(v.x * sc, v.y * sc, v.z * sc, v.w * sc);
}

// ---------- kernel 2: target inverse norms (one wave per row) ----------
__global__ void __launch_bounds__(256) norm_tgt_kernel(const float* __restrict__ tgt,
                                                       float* __restrict__ invt) {
    const int lid = threadIdx.x & 31, wid = threadIdx.x >> 5;
    const int row = blockIdx.x * 8 + wid;
    if (row >= TGT) return;
    const float4* r4 = (const float4*)(tgt + (size_t)row * DIMS);
    float ss = 0.0f;
    #pragma unroll
    for (int i = 0; i < 8; i++) {
        const float4 v = r4[lid + 32 * i];
        ss += v.x * v.x + v.y * v.y + v.z * v.z + v.w * v.w;
    }
    for (int off = 16; off > 0; off >>= 1) ss += __shfl_xor(ss, off, 32);
    if (lid == 0) invt[row] = 1.0f / fmaxf(sqrtf(ss), 1e-8f);
}

// ---------- kernel 3: WMMA GEMM (M=32) fused with per-slice top-4 ----------
__global__ void __launch_bounds__(256) gemm_topk_kernel(const float* __restrict__ srcn,
                                                        const float* __restrict__ tgt,
                                                        const float* __restrict__ invt,
                                                        float* __restrict__ pv,
                                                        int*   __restrict__ pi) {
    // 128 KB static LDS: two transposed A subtiles [k][m] (stride 16, conflict-free),
    // reused as merge scratch afterwards. (gfx1250: up to 320 KB/WGP.)
    __shared__ float smem[32768];
    const int tid  = threadIdx.x;
    const int lid  = tid & 31, wid = tid >> 5;
    const int nl   = lid & 15, half = lid >> 4;
    const int tile  = blockIdx.x;          // source tile (0..63), 32 rows each
    const int slice = blockIdx.y;          // target slice (0..7)
    const int row0  = tile * MT;
    const int sbase = slice * SLICE_LEN;
    const int send  = sbase + SLICE_LEN;

    // stage A transposed: rows 0..15 -> smem[k*16+m], rows 16..31 -> smem[16384 + k*16+m]
    for (int idx = tid; idx < MT * DIMS; idx += 256) {
        const int r = idx >> 10, c = idx & (DIMS - 1);
        const int base = (r < 16) ? 0 : 16384;
        smem[base + c * 16 + (r & 15)] = srcn[(size_t)(row0 + r) * DIMS + c];
    }
    __syncthreads();

    float tv[16][4]; int ti[16][4];
    #pragma unroll
    for (int rr = 0; rr < 16; rr++)
        #pragma unroll
        for (int j = 0; j < 4; j++) { tv[rr][j] = NEG_BIG; ti[rr][j] = -1; }

    const int kk = half * 2;
    const int ntiles = (SLICE_LEN + 15) >> 4;          // 391
    for (int t = wid; t < ntiles; t += 8) {            // uniform within a wave
        const int tbase = sbase + t * 16;
        const int gcol  = tbase + nl;
        const bool lane_ok = (gcol < send);
        const int ccol  = lane_ok ? gcol : (send - 1); // clamp: load real data, discard later
        const float* bp = tgt + (size_t)ccol * DIMS;
        // prefetch target row for this wave's NEXT tile (t+8 -> +128 rows)
        const int pcol  = (tbase + 128 + nl < send) ? (tbase + 128 + nl) : (send - 1);
        const float* bpn = tgt + (size_t)pcol * DIMS;

        v2f a0[2][8], a1[2][8], bb[2][8];
        auto load_batch = [&](int buf, int k0) {
            #pragma unroll
            for (int u = 0; u < 8; u++) {
                const int k = k0 + u * 4 + kk;
                a0[buf][u].x = smem[k * 16 + nl];
                a0[buf][u].y = smem[(k + 1) * 16 + nl];
                a1[buf][u].x = smem[16384 + k * 16 + nl];
                a1[buf][u].y = smem[16384 + (k + 1) * 16 + nl];
                bb[buf][u]   = *(const v2f*)(bp + k);
            }
        };

        v8f acc0 = {0.f,0.f,0.f,0.f,0.f,0.f,0.f,0.f};
        v8f acc1 = {0.f,0.f,0.f,0.f,0.f,0.f,0.f,0.f};
        load_batch(0, 0);
        // double-buffered: 32 k per batch, 32 batches total, step 2 batches/iter
        for (int kb = 0; kb < 32; kb += 2) {
            __builtin_prefetch(bpn + kb * 32 + kk, 0, 1);       // global_prefetch_b8
            load_batch(1, (kb + 1) * 32);
            #pragma unroll
            for (int u = 0; u < 8; u++) {
                acc0 = __builtin_amdgcn_wmma_f32_16x16x4_f32(
                    false, a0[0][u], false, bb[0][u], (short)0, acc0, false, false);
                acc1 = __builtin_amdgcn_wmma_f32_16x16x4_f32(
                    false, a1[0][u], false, bb[0][u], (short)0, acc1, false, false);
            }
            if (kb + 2 < 32) load_batch(0, (kb + 2) * 32);
            __builtin_prefetch(bpn + (kb + 1) * 32 + kk, 0, 1);
            #pragma unroll
            for (int u = 0; u < 8; u++) {
                acc0 = __builtin_amdgcn_wmma_f32_16x16x4_f32(
                    false, a0[1][u], false, bb[1][u], (short)0, acc0, false, false);
                acc1 = __builtin_amdgcn_wmma_f32_16x16x4_f32(
                    false, a1[1][u], false, bb[1][u], (short)0, acc1, false, false);
            }
        }

        const float sc = lane_ok ? invt[ccol] : 0.0f;
        #pragma unroll
        for (int r = 0; r < 8; r++) {
            const float v0 = lane_ok ? acc0[r] * sc : NEG_BIG;
            insert4(v0, gcol, tv[r], ti[r]);
            const float v1 = lane_ok ? acc1[r] * sc : NEG_BIG;
            insert4(v1, gcol, tv[r + 8], ti[r + 8]);
        }
    }

    __syncthreads();   // all waves done reading A region

    // dump per-lane candidates: cv[32 rows][128 cols][4], ci behind it (exactly 128 KB)
    float* cv = smem;
    int*   ci = (int*)smem + 16384;
    #pragma unroll
    for (int rr = 0; rr < 16; rr++) {
        const int m   = (rr < 8) ? (rr + 8 * half) : (16 + (rr - 8) + 8 * half);
        const int col = wid * 16 + nl;
        #pragma unroll
        for (int j = 0; j < 4; j++) {
            cv[(m * 128 + col) * 4 + j] = tv[rr][j];
            ci[(m * 128 + col) * 4 + j] = ti[rr][j];
        }
    }
    __syncthreads();

    // wave `wid` merges rows wid*4 .. wid*4+3: 512 candidates -> top-4 each
    for (int mm = 0; mm < 4; mm++) {
        const int m = wid * 4 + mm;
        float mv[4]; int mi[4];
        #pragma unroll
        for (int j = 0; j < 4; j++) { mv[j] = NEG_BIG; mi[j] = -1; }
        #pragma unroll
        for (int tc = 0; tc < 4; tc++) {
            const int col = lid + 32 * tc;
            #pragma unroll
            for (int j = 0; j < 4; j++)
                insert4(cv[(m * 128 + col) * 4 + j], ci[(m * 128 + col) * 4 + j], mv, mi);
        }
        float rv[4]; int ri[4];
        #pragma unroll
        for (int j = 0; j < 4; j++) {
            float bv = mv[0]; int bi = mi[0];
            for (int off = 16; off > 0; off >>= 1) {
                const float ov = __shfl_xor(bv, off, 32);
                const int   oi = __shfl_xor(bi, off, 32);
                if (ov > bv || (ov == bv && (unsigned)oi < (unsigned)bi)) { bv = ov; bi = oi; }
            }
            rv[j] = bv; ri[j] = bi;
            if (mi[0] == bi) {   // pop (target indices unique per row)
                mv[0] = mv[1]; mi[0] = mi[1];
                mv[1] = mv[2]; mi[1] = mi[2];
                mv[2] = mv[3]; mi[2] = mi[3];
                mv[3] = NEG_BIG; mi[3] = -1;
            }
        }
        if (lid == 0) {
            const int grow = row0 + m;
            #pragma unroll
            for (int j = 0; j < 4; j++) {
                pv[((size_t)slice * SRC + grow) * 4 + j] = rv[j];
                pi[((size_t)slice * SRC + grow) * 4 + j] = ri[j];
            }
        }
    }
}

// ---------- kernel 4: merge slice partials, gather + average ----------
__global__ void __launch_bounds__(256) merge_out_kernel(const float* __restrict__ tgt,
                                                        const float* __restrict__ pv,
                                                        const int*   __restrict__ pi,
                                                        float* __restrict__ out) {
    const int lid = threadIdx.x & 31, wid = threadIdx.x >> 5;
    const int m = blockIdx.x * 8 + wid;          // source row, 2048 total
    // 32 candidates (8 slices x 4) -> one per lane
    float v = pv[((size_t)(lid >> 2) * SRC + m) * 4 + (lid & 3)];
    int   i = pi[((size_t)(lid >> 2) * SRC + m) * 4 + (lid & 3)];
    int idxs[4];
    #pragma unroll
    for (int j = 0; j < 4; j++) {
        float bv = v; int bi = i;
        for (int off = 16; off > 0; off >>= 1) {
            const float ov = __shfl_xor(bv, off, 32);
            const int   oi = __shfl_xor(bi, off, 32);
            if (ov > bv || (ov == bv && (unsigned)oi < (unsigned)bi)) { bv = ov; bi = oi; }
        }
        idxs[j] = bi;
        if (i == bi) v = NEG_BIG;    // pop (indices unique across slices)
    }
    const float4* t0 = (const float4*)(tgt + (size_t)idxs[0] * DIMS);
    const float4* t1 = (const float4*)(tgt + (size_t)idxs[1] * DIMS);
    const float4* t2 = (const float4*)(tgt + (size_t)idxs[2] * DIMS);
    const float4* t3 = (const float4*)(tgt + (size_t)idxs[3] * DIMS);
    float4* o4 = (float4*)(out + (size_t)m * DIMS);
    #pragma unroll
    for (int q = 0; q < 8; q++) {
        const int d = lid + 32 * q;
        const float4 a = t0[d], b = t1[d], c = t2[d], e = t3[d];
        o4[d] = make_float4(0.25f * (a.x + b.x + c.x + e.x),
                            0.25f * (a.y + b.y + c.y + e.y),
                            0.25f * (a.z + b.z + c.z + e.z),
                            0.25f * (a.w + b.w + c.w + e.w));
    }
}

// ---------- launch ----------
extern "C" void kernel_launch(void* const* d_in, const int* in_sizes, int n_in,
                              void* d_out, int out_size, void* d_ws, size_t ws_size,
                              hipStream_t stream) {
    const float* src = (const float*)d_in[0];   // [2048,1024] f32
    const float* tgt = (const float*)d_in[1];   // [50000,1024] f32
    float* out = (float*)d_out;                 // [2048,1024] f32
    char* ws = (char*)d_ws;

    // workspace layout (bytes)
    float* srcn = (float*)(ws);                     // 2048*1024*4 = 8388608
    float* invt = (float*)(ws + 8388608);           // 50000*4     -> pad 204800
    float* pv   = (float*)(ws + 8593408);           // 8*2048*4*4  = 262144
    int*   pi   = (int*)  (ws + 8855552);           // 262144  (total ~8.7 MiB)

    norm_src_kernel<<<SRC, 256, 0, stream>>>(src, srcn);
    norm_tgt_kernel<<<TGT / 8, 256, 0, stream>>>(tgt, invt);
    gemm_topk_kernel<<<dim3(SRC / MT, SLICES), 256, 0, stream>>>(srcn, tgt, invt, pv, pi);
    merge_out_kernel<<<SRC / 8, 256, 0, stream>>>(tgt, pv, pi, out);
}